// DGCNN_78761110274728
// MI455X (gfx1250) — compile-verified
//
#include <hip/hip_runtime.h>
#include <math.h>

typedef float v2f __attribute__((ext_vector_type(2)));
typedef float v8f __attribute__((ext_vector_type(8)));

constexpr int Bsz  = 8;
constexpr int Npts = 2048;
constexpr int KNN  = 32;
constexpr int C1   = 48;
constexpr int C2   = 96;
constexpr int CC   = 144;   // C1 + C2
constexpr int FD   = 96;
constexpr int GN_G = 12;
constexpr float BN_EPS = 1e-5f;
constexpr float GN_EPS = 1e-5f;

// ---------------------------------------------------------------------------
// 1) kNN + gather. One thread per query point; all 2048 points of the batch
//    cached in LDS. Register-resident sorted top-32 (ascending distance,
//    ties -> lower index first, matching jax.lax.top_k(-d)).
//    Writes gathered neighbor coords: feat[b][n][k][2].
// ---------------------------------------------------------------------------
__global__ __launch_bounds__(256)
void knn_gather_kernel(const float* __restrict__ x, const float* __restrict__ y,
                       float* __restrict__ featx, float* __restrict__ featy) {
  const int b = blockIdx.y;
  const int z = blockIdx.z;
  const float* pts = (z ? y : x) + (size_t)b * Npts * 2;
  float* feat = (z ? featy : featx);

  __shared__ float sx[Npts], sy[Npts], sq[Npts];
  for (int i = threadIdx.x; i < Npts; i += 256) {
    float px = pts[2 * i], py = pts[2 * i + 1];
    sx[i] = px; sy[i] = py; sq[i] = px * px + py * py;
  }
  __syncthreads();

  const int n = blockIdx.x * 256 + threadIdx.x;
  const float qx = sx[n], qy = sy[n], qs = sq[n];

  float bd[KNN]; int bi[KNN];
#pragma unroll
  for (int i = 0; i < KNN; ++i) { bd[i] = INFINITY; bi[i] = 0x7fffffff; }

  for (int j = 0; j < Npts; ++j) {
    float d = qs + sq[j] - 2.0f * (qx * sx[j] + qy * sy[j]);
    if (d < bd[KNN - 1]) {
      float pd = d; int pj = j;
#pragma unroll
      for (int i = 0; i < KNN; ++i) {
        bool lt = (pd < bd[i]) || (pd == bd[i] && pj < bi[i]);
        float od = bd[i]; int oi = bi[i];
        bd[i] = lt ? pd : od;  bi[i] = lt ? pj : oi;
        pd    = lt ? od : pd;  pj    = lt ? oi : pj;
      }
    }
  }

  float* F = feat + (((size_t)b * Npts + n) * KNN) * 2;
#pragma unroll
  for (int k = 0; k < KNN; ++k) {
    int j = bi[k];
    F[2 * k]     = sx[j];
    F[2 * k + 1] = sy[j];
  }
}

// ---------------------------------------------------------------------------
// 2) Branch kernel: per block = 8 points x 32 neighbors = 256 "pixels".
//    h1 (48ch) computed per-thread into LDS; h2 (96ch) via f32 WMMA
//    (16x16x4) with bn2+relu fused on the accumulator; m2w-weighted
//    k-reduction via shfl_xor butterflies + LDS atomicAdd.
//    Output x3[b][n][144] = [x1(48) | x2(96)].
// ---------------------------------------------------------------------------
__global__ __launch_bounds__(256)
void branch_kernel(const float* __restrict__ featx, const float* __restrict__ featy,
                   const float* __restrict__ W1,
                   const float* __restrict__ bn1_g, const float* __restrict__ bn1_b,
                   const float* __restrict__ bn1_m, const float* __restrict__ bn1_v,
                   const float* __restrict__ m1w,  const float* __restrict__ m1b,
                   const float* __restrict__ W2,
                   const float* __restrict__ bn2_g, const float* __restrict__ bn2_b,
                   const float* __restrict__ bn2_m, const float* __restrict__ bn2_v,
                   const float* __restrict__ m2w,  const float* __restrict__ m2b,
                   float* __restrict__ x3x, float* __restrict__ x3y) {
  const int b = blockIdx.y, z = blockIdx.z;
  const float* feat = z ? featy : featx;
  float* x3 = z ? x3y : x3x;
  const int n0  = blockIdx.x * 8;
  const int tid = threadIdx.x;

  __shared__ float h1s[C1 * 256];          // 48 KB
  __shared__ float x2s[8 * C2];            //  3 KB
  __shared__ float w1s[2 * C1], sc1[C1], sh1[C1], sc2[C2], sh2[C2];
  __shared__ float m1ws[KNN], m2ws[KNN];

  if (tid < C1) {
    float s = bn1_g[tid] * rsqrtf(bn1_v[tid] + BN_EPS);
    sc1[tid] = s; sh1[tid] = bn1_b[tid] - bn1_m[tid] * s;
  }
  if (tid < C2) {
    float s = bn2_g[tid] * rsqrtf(bn2_v[tid] + BN_EPS);
    sc2[tid] = s; sh2[tid] = bn2_b[tid] - bn2_m[tid] * s;
  }
  if (tid < 2 * C1) w1s[tid] = W1[tid];
  if (tid < KNN) { m1ws[tid] = m1w[tid]; m2ws[tid] = m2w[tid]; }
  for (int i = tid; i < 8 * C2; i += 256) x2s[i] = 0.0f;

  // h1: one pixel per thread
  {
    const int nl = tid >> 5, k = tid & 31;
    const float* fp = feat + (((size_t)b * Npts + n0 + nl) * KNN + k) * 2;
    const float fx_ = fp[0], fy_ = fp[1];
    __syncthreads();
#pragma unroll
    for (int c = 0; c < C1; ++c) {
      float zv = fx_ * w1s[2 * c] + fy_ * w1s[2 * c + 1];
      h1s[c * 256 + tid] = fmaxf(zv * sc1[c] + sh1[c], 0.0f);
    }
  }
  __syncthreads();

  // h2 via WMMA f32 16x16x4, fused bn2+relu and weighted k-reduction
  const int wave = tid >> 5, lane = tid & 31;
  const int half = lane >> 4, l16 = lane & 15;
#pragma unroll
  for (int ot = 0; ot < 6; ++ot) {
    const int o0 = ot * 16;
    v2f A[12];
#pragma unroll
    for (int cs = 0; cs < 12; ++cs) {
      const int c0 = cs * 4 + 2 * half;
      A[cs] = *(const v2f*)(W2 + (size_t)(o0 + l16) * C1 + c0);
    }
#pragma unroll
    for (int gi = 0; gi < 2; ++gi) {
      const int g  = wave + 8 * gi;     // pixel group 0..15 (16 pixels each)
      const int p0 = g * 16;
      v8f acc = {};
#pragma unroll
      for (int cs = 0; cs < 12; ++cs) {
        const int c0 = cs * 4 + 2 * half;
        v2f Bm;
        Bm.x = h1s[(c0 + 0) * 256 + p0 + l16];
        Bm.y = h1s[(c0 + 1) * 256 + p0 + l16];
        acc = __builtin_amdgcn_wmma_f32_16x16x4_f32(
            false, A[cs], false, Bm, (short)0, acc, false, false);
      }
      const int nl = g >> 1;                 // local point 0..7
      const int k  = (g & 1) * 16 + l16;     // neighbor slot 0..31
      const float mw = m2ws[k];
#pragma unroll
      for (int i = 0; i < 8; ++i) {
        const int c = o0 + i + 8 * half;
        float h2 = fmaxf(acc[i] * sc2[c] + sh2[c], 0.0f);
        float part = h2 * mw;
        part += __shfl_xor(part, 1, 32);
        part += __shfl_xor(part, 2, 32);
        part += __shfl_xor(part, 4, 32);
        part += __shfl_xor(part, 8, 32);
        if (l16 == 0) atomicAdd(&x2s[nl * C2 + c], part);
      }
    }
  }
  __syncthreads();

  const float m1b0 = m1b[0], m2b0 = m2b[0];
  for (int idx = tid; idx < 8 * C1; idx += 256) {
    const int nl = idx / C1, c = idx % C1;
    float s = 0.0f;
#pragma unroll
    for (int k = 0; k < KNN; ++k) s += h1s[c * 256 + nl * KNN + k] * m1ws[k];
    x3[((size_t)b * Npts + n0 + nl) * CC + c] = s + m1b0;
  }
  for (int idx = tid; idx < 8 * C2; idx += 256) {
    const int nl = idx / C2, c = idx % C2;
    x3[((size_t)b * Npts + n0 + nl) * CC + C1 + c] = x2s[idx] + m2b0;
  }
}

// ---------------------------------------------------------------------------
// 3) Max-pool over N per (b, channel) for both branches.
// ---------------------------------------------------------------------------
__global__ __launch_bounds__(256)
void pool_kernel(const float* __restrict__ x3x, const float* __restrict__ x3y,
                 float* __restrict__ poolx, float* __restrict__ pooly) {
  const int c = blockIdx.x, b = blockIdx.y, z = blockIdx.z;
  const float* x3 = z ? x3y : x3x;
  float m = -INFINITY;
  for (int n = threadIdx.x; n < Npts; n += 256)
    m = fmaxf(m, x3[((size_t)b * Npts + n) * CC + c]);
  __shared__ float red[256];
  red[threadIdx.x] = m; __syncthreads();
  for (int s = 128; s > 0; s >>= 1) {
    if (threadIdx.x < s) red[threadIdx.x] = fmaxf(red[threadIdx.x], red[threadIdx.x + s]);
    __syncthreads();
  }
  if (threadIdx.x == 0) (z ? pooly : poolx)[b * CC + c] = red[0];
}

// ---------------------------------------------------------------------------
// 4) Fold the broadcast half of prepool into a per-(b,o) bias:
//    bias2[b][o] = bp[o] + sum_c Wp[o][144+c] * pool_other[b][c]
//    (fx uses pool_y, fy uses pool_x)
// ---------------------------------------------------------------------------
__global__ __launch_bounds__(FD)
void bias2_kernel(const float* __restrict__ Wp, const float* __restrict__ bp,
                  const float* __restrict__ poolx, const float* __restrict__ pooly,
                  float* __restrict__ b2x, float* __restrict__ b2y) {
  const int b = blockIdx.x, z = blockIdx.y;
  const int o = threadIdx.x;
  const float* pool = z ? poolx : pooly;
  float s = bp[o];
  for (int c = 0; c < CC; ++c) s += Wp[(size_t)o * 288 + CC + c] * pool[b * CC + c];
  (z ? b2y : b2x)[b * FD + o] = s;
}

// ---------------------------------------------------------------------------
// 5) prepool GEMM (96 x 144) over N pixels via WMMA f32 16x16x4, bias init
//    from bias2 in the accumulator. Writes h to d_out (normalized in place
//    later).
// ---------------------------------------------------------------------------
__global__ __launch_bounds__(256)
void prepool_kernel(const float* __restrict__ Wp,
                    const float* __restrict__ x3x, const float* __restrict__ x3y,
                    const float* __restrict__ b2x, const float* __restrict__ b2y,
                    float* __restrict__ out) {
  const int b = blockIdx.y, z = blockIdx.z;
  const float* x3    = z ? x3y : x3x;
  const float* bias2 = z ? b2y : b2x;
  float* o_ = out + (size_t)z * Bsz * Npts * FD;

  __shared__ float wps[FD * CC];           // 54 KB: Wp[:, :144]
  for (int i = threadIdx.x; i < FD * CC; i += 256) {
    int o = i / CC, c = i % CC;
    wps[i] = Wp[(size_t)o * 288 + c];
  }
  __syncthreads();

  const int wave = threadIdx.x >> 5, lane = threadIdx.x & 31;
  const int half = lane >> 4, l16 = lane & 15;
  const int npt = blockIdx.x * 128 + wave * 16 + l16;
  const float* xrow = x3 + ((size_t)b * Npts + npt) * CC;

#pragma unroll
  for (int ot = 0; ot < 6; ++ot) {
    const int o0 = ot * 16;
    v8f acc;
#pragma unroll
    for (int i = 0; i < 8; ++i) acc[i] = bias2[b * FD + o0 + i + 8 * half];
#pragma unroll
    for (int cs = 0; cs < 36; ++cs) {
      const int c0 = cs * 4 + 2 * half;
      v2f A  = *(const v2f*)(wps + (size_t)(o0 + l16) * CC + c0);
      v2f Bm = *(const v2f*)(xrow + c0);
      acc = __builtin_amdgcn_wmma_f32_16x16x4_f32(
          false, A, false, Bm, (short)0, acc, false, false);
    }
    float* orow = o_ + ((size_t)b * Npts + npt) * FD + o0 + 8 * half;
#pragma unroll
    for (int i = 0; i < 8; ++i) orow[i] = acc[i];
  }
}

// ---------------------------------------------------------------------------
// 6) GroupNorm statistics per (branch, batch, group): mean + rstd over
//    8 channels x 2048 points.
// ---------------------------------------------------------------------------
__global__ __launch_bounds__(256)
void gn_stats_kernel(const float* __restrict__ out,
                     float* __restrict__ gnm, float* __restrict__ gnr) {
  const int g = blockIdx.x, b = blockIdx.y, z = blockIdx.z;
  const float* o_ = out + (size_t)z * Bsz * Npts * FD;
  float s = 0.0f, s2 = 0.0f;
  for (int n = threadIdx.x; n < Npts; n += 256) {
    const float* row = o_ + ((size_t)b * Npts + n) * FD + g * 8;
#pragma unroll
    for (int j = 0; j < 8; ++j) { float v = row[j]; s += v; s2 += v * v; }
  }
  __shared__ float r1[256], r2[256];
  r1[threadIdx.x] = s; r2[threadIdx.x] = s2; __syncthreads();
  for (int st = 128; st > 0; st >>= 1) {
    if (threadIdx.x < st) {
      r1[threadIdx.x] += r1[threadIdx.x + st];
      r2[threadIdx.x] += r2[threadIdx.x + st];
    }
    __syncthreads();
  }
  if (threadIdx.x == 0) {
    const float inv = 1.0f / (8.0f * Npts);
    float mean = r1[0] * inv;
    float var  = r2[0] * inv - mean * mean;
    int idx = (z * Bsz + b) * GN_G + g;
    gnm[idx] = mean;
    gnr[idx] = rsqrtf(var + GN_EPS);
  }
}

// ---------------------------------------------------------------------------
// 7) Apply GroupNorm + relu + per-point L2 normalization, in place in d_out.
// ---------------------------------------------------------------------------
__global__ __launch_bounds__(256)
void gn_apply_kernel(float* __restrict__ out,
                     const float* __restrict__ gn_g, const float* __restrict__ gn_b,
                     const float* __restrict__ gnm, const float* __restrict__ gnr) {
  const int b = blockIdx.y, z = blockIdx.z;
  const int n = blockIdx.x * 256 + threadIdx.x;
  __shared__ float gs[FD], gb[FD], ms[GN_G], rs[GN_G];
  if (threadIdx.x < FD) { gs[threadIdx.x] = gn_g[threadIdx.x]; gb[threadIdx.x] = gn_b[threadIdx.x]; }
  if (threadIdx.x < GN_G) {
    int idx = (z * Bsz + b) * GN_G + threadIdx.x;
    ms[threadIdx.x] = gnm[idx]; rs[threadIdx.x] = gnr[idx];
  }
  __syncthreads();
  float* row = out + (size_t)z * Bsz * Npts * FD + ((size_t)b * Npts + n) * FD;
  float v[FD];
  float ss = 0.0f;
#pragma unroll
  for (int c = 0; c < FD; ++c) {
    const int g = c >> 3;
    float t = (row[c] - ms[g]) * rs[g] * gs[c] + gb[c];
    t = fmaxf(t, 0.0f);
    v[c] = t; ss += t * t;
  }
  const float nrm = sqrtf(ss);
#pragma unroll
  for (int c = 0; c < FD; ++c) row[c] = v[c] / nrm;
}

// ---------------------------------------------------------------------------
extern "C" void kernel_launch(void* const* d_in, const int* in_sizes, int n_in,
                              void* d_out, int out_size, void* d_ws, size_t ws_size,
                              hipStream_t stream) {
  (void)in_sizes; (void)n_in; (void)out_size; (void)ws_size;
  const float* x     = (const float*)d_in[0];
  const float* y     = (const float*)d_in[1];
  const float* W1    = (const float*)d_in[2];
  const float* bn1_g = (const float*)d_in[3];
  const float* bn1_b = (const float*)d_in[4];
  const float* bn1_m = (const float*)d_in[5];
  const float* bn1_v = (const float*)d_in[6];
  const float* m1w   = (const float*)d_in[7];
  const float* m1b   = (const float*)d_in[8];
  const float* W2    = (const float*)d_in[9];
  const float* bn2_g = (const float*)d_in[10];
  const float* bn2_b = (const float*)d_in[11];
  const float* bn2_m = (const float*)d_in[12];
  const float* bn2_v = (const float*)d_in[13];
  const float* m2w   = (const float*)d_in[14];
  const float* m2b   = (const float*)d_in[15];
  const float* Wp    = (const float*)d_in[16];
  const float* bp    = (const float*)d_in[17];
  const float* gn_g  = (const float*)d_in[18];
  const float* gn_b  = (const float*)d_in[19];
  float* out = (float*)d_out;
  float* ws  = (float*)d_ws;

  const size_t FEAT_SZ = (size_t)Bsz * Npts * KNN * 2;  // 1,048,576 f
  const size_t X3_SZ   = (size_t)Bsz * Npts * CC;       // 2,359,296 f
  const size_t POOL_SZ = (size_t)Bsz * CC;
  const size_t B2_SZ   = (size_t)Bsz * FD;
  const size_t GN_SZ   = (size_t)2 * Bsz * GN_G;

  float* featx = ws;
  float* featy = featx + FEAT_SZ;
  float* x3x   = featy + FEAT_SZ;
  float* x3y   = x3x + X3_SZ;
  float* poolx = x3y + X3_SZ;
  float* pooly = poolx + POOL_SZ;
  float* b2x   = pooly + POOL_SZ;
  float* b2y   = b2x + B2_SZ;
  float* gnm   = b2y + B2_SZ;
  float* gnr   = gnm + GN_SZ;

  knn_gather_kernel<<<dim3(Npts / 256, Bsz, 2), 256, 0, stream>>>(x, y, featx, featy);
  branch_kernel<<<dim3(Npts / 8, Bsz, 2), 256, 0, stream>>>(
      featx, featy, W1, bn1_g, bn1_b, bn1_m, bn1_v, m1w, m1b,
      W2, bn2_g, bn2_b, bn2_m, bn2_v, m2w, m2b, x3x, x3y);
  pool_kernel<<<dim3(CC, Bsz, 2), 256, 0, stream>>>(x3x, x3y, poolx, pooly);
  bias2_kernel<<<dim3(Bsz, 2, 1), FD, 0, stream>>>(Wp, bp, poolx, pooly, b2x, b2y);
  prepool_kernel<<<dim3(Npts / 128, Bsz, 2), 256, 0, stream>>>(Wp, x3x, x3y, b2x, b2y, out);
  gn_stats_kernel<<<dim3(GN_G, Bsz, 2), 256, 0, stream>>>(out, gnm, gnr);
  gn_apply_kernel<<<dim3(Npts / 256, Bsz, 2), 256, 0, stream>>>(out, gn_g, gn_b, gnm, gnr);
}